// GraphReshape_16338055594072
// MI455X (gfx1250) — compile-verified
//
#include <hip/hip_runtime.h>
#include <hip/hip_bf16.h>
#include <math.h>

typedef float v2f __attribute__((ext_vector_type(2)));
typedef float v8f __attribute__((ext_vector_type(8)));

#define XD 128
#define HD 128
#define WSTRIDE 132   // padded LDS row stride (words): bank stride 4 -> conflict-free

// ---------------- zero the aggregation scratch ----------------
__global__ void zero_ws(float4* __restrict__ p, int n4) {
    int i = blockIdx.x * blockDim.x + threadIdx.x;
    float4 z = {0.f, 0.f, 0.f, 0.f};
    for (; i < n4; i += gridDim.x * blockDim.x) p[i] = z;
}

// ---------------- COO edge scatter: agg[row] += val * x[col] ----------------
// One wave32 per edge; lane covers 4 consecutive floats (128 = 32*4).
// x (51.2 MB) and agg (51.2 MB) both fit in the 192 MB L2, so both the
// gather and the f32 atomics resolve on-chip (global_atomic_add_f32).
__global__ void edge_scatter(const float* __restrict__ x,
                             const int* __restrict__ rows,
                             const int* __restrict__ cols,
                             const float* __restrict__ vals,
                             float* __restrict__ agg, int E) {
    int wave = (blockIdx.x * blockDim.x + threadIdx.x) >> 5;
    int lane = threadIdx.x & 31;
    if (wave >= E) return;
    int r = rows[wave];          // uniform per wave -> scalarized
    int c = cols[wave];
    float v = vals[wave];
    float4 xv = ((const float4*)(x + (long)c * XD))[lane];   // coalesced 512B/wave
    float* dst = agg + (long)r * XD + lane * 4;
    unsafeAtomicAdd(dst + 0, v * xv.x);
    unsafeAtomicAdd(dst + 1, v * xv.y);
    unsafeAtomicAdd(dst + 2, v * xv.z);
    unsafeAtomicAdd(dst + 3, v * xv.w);
}

// ---------------- encode: softmax(prelu(in @ W^T + b)) ----------------
// One wave per 16-row tile; 8 waves per block. W is staged once per block in
// LDS with a padded stride so the per-k ds_load_b64 B-fragment reads are
// bank-conflict-free across all 32 lanes.
// V_WMMA_F32_16X16X4_F32 fragment layouts (wave32):
//   A (16x4):  lanes 0-15 hold M=0..15; VGPR0/1 = K{0,1} (lo half) or K{2,3} (hi half)
//   B (4x16):  lanes 0-15 hold N=0..15; VGPR0/1 = K{0,1} (lo half) or K{2,3} (hi half)
//   C/D (16x16): VGPR r = row r (lanes 0-15, N=lane) / row 8+r (lanes 16-31, N=lane-16)
__global__ void __launch_bounds__(256)
encode_wmma(const float* __restrict__ in,
            const float* __restrict__ W,
            const float* __restrict__ bias,
            const float* __restrict__ aptr,
            float* __restrict__ out,
            int N, int ntiles) {
    __shared__ float Wl[HD * WSTRIDE];     // 128 rows x 132 words = 66 KB

    // Cooperative W load: float4 global reads, ds_store_b128 (528-byte rows
    // keep 16B alignment). All 256 threads participate before any exit.
    {
        const float4* Wv = (const float4*)W;
        for (int idx4 = threadIdx.x; idx4 < (HD * XD) / 4; idx4 += 256) {
            int n  = idx4 >> 5;            // row (128 floats = 32 float4)
            int k4 = (idx4 & 31) << 2;     // col start
            *(float4*)(&Wl[n * WSTRIDE + k4]) = Wv[idx4];
        }
    }
    __syncthreads();

    int tile = blockIdx.x * (blockDim.x >> 5) + (threadIdx.x >> 5);
    if (tile >= ntiles) return;
    int lane = threadIdx.x & 31;
    int half = lane >> 4;        // K-half for A/B frags, M-half for C/D
    int l    = lane & 15;
    int row0 = tile * 16;
    float alpha = aptr[0];

    v8f acc[8];
    #pragma unroll
    for (int c = 0; c < 8; ++c) acc[c] = (v8f)(0.f);

    int arow_idx = row0 + l;
    if (arow_idx >= N) arow_idx = N - 1;                     // tail clamp (N%16==0 normally)
    const float* arow = in + (long)arow_idx * XD + 2 * half; // 8B-aligned

    // Lane's LDS base for B fragments: row (c*16+l), cols k+2*half.
    const float* wl_base = &Wl[(long)l * WSTRIDE + 2 * half];

    v2f af = *(const v2f*)(arow);                            // prefetch k=0
    for (int k = 0; k < XD; k += 4) {
        // Issue all 8 independent B-fragment LDS loads up front
        v2f bf[8];
        #pragma unroll
        for (int c = 0; c < 8; ++c)
            bf[c] = *(const v2f*)(wl_base + (c * 16) * WSTRIDE + k);
        v2f af_cur = af;
        if (k + 4 < XD) af = *(const v2f*)(arow + k + 4);    // prefetch next A frag
        #pragma unroll
        for (int c = 0; c < 8; ++c)
            acc[c] = __builtin_amdgcn_wmma_f32_16x16x4_f32(
                false, af_cur, false, bf[c], (short)0, acc[c], false, false);
    }

    // bias values for this lane's N positions across the 8 column tiles
    float bn[8];
    #pragma unroll
    for (int c = 0; c < 8; ++c) bn[c] = bias[c * 16 + l];

    // Per accumulator row r: values of output row (row0 + r + 8*half) live in
    // acc[c][r] across lanes l=0..15 of this half. xor-shuffles with mask<16
    // stay within the half, giving the 16-lane rowwise reduction we need.
    #pragma unroll
    for (int r = 0; r < 8; ++r) {
        float v[8];
        float m = -INFINITY;
        #pragma unroll
        for (int c = 0; c < 8; ++c) {
            float t = acc[c][r] + bn[c];
            t = t > 0.f ? t : alpha * t;                     // PReLU
            v[c] = t;
            m = fmaxf(m, t);
        }
        #pragma unroll
        for (int off = 1; off < 16; off <<= 1) m = fmaxf(m, __shfl_xor(m, off, 32));
        float s = 0.f;
        #pragma unroll
        for (int c = 0; c < 8; ++c) { v[c] = __expf(v[c] - m); s += v[c]; }
        #pragma unroll
        for (int off = 1; off < 16; off <<= 1) s += __shfl_xor(s, off, 32);
        float inv = 1.f / s;
        int row = row0 + r + 8 * half;
        if (row < N) {
            #pragma unroll
            for (int c = 0; c < 8; ++c)
                out[(long)row * HD + c * 16 + l] = v[c] * inv;
        }
    }
}

extern "C" void kernel_launch(void* const* d_in, const int* in_sizes, int n_in,
                              void* d_out, int out_size, void* d_ws, size_t ws_size,
                              hipStream_t stream) {
    const float* x    = (const float*)d_in[0];
    const int*   rows = (const int*)d_in[1];
    const int*   cols = (const int*)d_in[2];
    const float* vals = (const float*)d_in[3];
    const float* W    = (const float*)d_in[4];
    const float* bias = (const float*)d_in[5];
    const float* a    = (const float*)d_in[6];
    float* out = (float*)d_out;
    float* agg = (float*)d_ws;               // N*128 floats = 51.2 MB scratch

    int N = in_sizes[0] / XD;                // 100000
    int E = in_sizes[1];                     // 3.2M

    // 1) zero agg
    int n4 = (N * XD) / 4;
    zero_ws<<<2048, 256, 0, stream>>>((float4*)agg, n4);

    // 2) edge scatter (one wave per edge, 8 waves per block)
    int eblk = (E + 7) / 8;
    edge_scatter<<<eblk, 256, 0, stream>>>(x, rows, cols, vals, agg, E);

    // 3) encode both matrices with f32 WMMA
    int ntiles = (N + 15) / 16;              // 6250
    int gblk = (ntiles + 7) / 8;             // 8 waves per block
    encode_wmma<<<gblk, 256, 0, stream>>>(x,   W, bias, a, out,                 N, ntiles);
    encode_wmma<<<gblk, 256, 0, stream>>>(agg, W, bias, a, out + (long)N * HD,  N, ntiles);
}